// ResMPNN_26534307954801
// MI455X (gfx1250) — compile-verified
//
#include <hip/hip_runtime.h>
#include <hip/hip_bf16.h>
#include <math.h>

// ---------------------------------------------------------------------------
// ResMPNN layer for MI455X (gfx1250, wave32, WMMA).
// B=8, L=512, K=32, D=E=128.  All GEMMs via V_WMMA_F32_16X16X4_F32 (fp32).
// ---------------------------------------------------------------------------

#define BB 8
#define LL 512
#define KK 32
#define DD 128
#define EE 128

// LDS strides: chosen coprime-ish with 64 banks (387 % 64 = 3, 131 % 64 = 3)
// so column reads (A fragments) and tile writes are conflict-free.
#define XS 387
#define YS 131

typedef float v2f __attribute__((ext_vector_type(2)));
typedef float v8f __attribute__((ext_vector_type(8)));

__device__ __forceinline__ float gelu_exact(float x) {
    // 0.5 * x * (1 + erf(x / sqrt(2)))  -- matches jax.nn.gelu(approximate=False)
    return 0.5f * x * (1.0f + erff(x * 0.70710678118654752f));
}

// Y[0..31][0..127] = gelu(X[0..31][0..Kdim-1] @ W[Kdim][128] + bias)
// 8 waves: wave w owns output columns n0 = 16*w; it computes both 16-row
// M tiles (rows 0-15 and 16-31), reusing the B (weights) fragment.
//
// V_WMMA_F32_16X16X4_F32 fragment layouts (ISA 7.12.2, 32-bit, wave32):
//   A 16x4 : lanes 0-15 -> M=lane, {v0,v1} = K{0,1}; lanes 16-31 -> K{2,3}
//   B 4x16 : lanes 0-15 -> N=lane, {v0,v1} = K{0,1}; lanes 16-31 -> K{2,3}
//   C 16x16: VGPR r, lanes 0-15 -> (M=r, N=lane); lanes 16-31 -> (M=r+8)
__device__ __forceinline__ void mlp_layer(const float* __restrict__ W,
                                          const float* __restrict__ bias,
                                          const float* Xl, int xStride, int Kdim,
                                          float* Yl, int yStride,
                                          int lane, int wave) {
    const int n0 = wave * 16;
    const int hf = lane >> 4;    // half-wave select (K pair)
    const int lr = lane & 15;    // row/col within half

    v8f c0 = {};                 // rows 0..15  tile
    v8f c1 = {};                 // rows 16..31 tile

    for (int k = 0; k < Kdim; k += 4) {
        const int kk = k + 2 * hf;

        v2f bfrag;
        bfrag.x = W[(size_t)kk * DD + n0 + lr];
        bfrag.y = W[(size_t)(kk + 1) * DD + n0 + lr];

        v2f a0, a1;
        a0.x = Xl[lr * xStride + kk];
        a0.y = Xl[lr * xStride + kk + 1];
        a1.x = Xl[(16 + lr) * xStride + kk];
        a1.y = Xl[(16 + lr) * xStride + kk + 1];

        c0 = __builtin_amdgcn_wmma_f32_16x16x4_f32(false, a0, false, bfrag,
                                                   (short)0, c0, false, false);
        c1 = __builtin_amdgcn_wmma_f32_16x16x4_f32(false, a1, false, bfrag,
                                                   (short)0, c1, false, false);
    }

    const float bn = bias[n0 + lr];
#pragma unroll
    for (int r = 0; r < 8; ++r) {
        const int row0 = r + 8 * hf;
        const int row1 = 16 + r + 8 * hf;
        Yl[row0 * yStride + n0 + lr] = gelu_exact(c0[r] + bn);
        Yl[row1 * yStride + n0 + lr] = gelu_exact(c1[r] + bn);
    }
}

// ---------------------------------------------------------------------------
// Kernel 1: message MLP (384->128->128->128, gelu) + masked-mean aggregation.
// One block per node (b,l). Writes h_tmp[b,l,:] = (res + msg_mean) * mask.
// ---------------------------------------------------------------------------
__global__ __launch_bounds__(256)
void msg_kernel(const float* __restrict__ res_emb,
                const float* __restrict__ edge_emb,
                const int*   __restrict__ edge_index,
                const float* __restrict__ mask,
                const float* __restrict__ W1, const float* __restrict__ b1,
                const float* __restrict__ W2, const float* __restrict__ b2,
                const float* __restrict__ W3, const float* __restrict__ b3,
                float* __restrict__ h_tmp) {
    __shared__ float Xs[KK * XS];
    __shared__ float Ysh[KK * YS];
    __shared__ float hc[DD];
    __shared__ int   sidx[KK];
    __shared__ float svalid[KK];
    __shared__ float snmask[KK];
    __shared__ float snv;

    const int tid  = threadIdx.x;
    const int b    = blockIdx.x >> 9;       // / L
    const int l    = blockIdx.x & (LL - 1); // % L
    const int node = b * LL + l;
    const float mC = mask[node];

    if (tid < DD) hc[tid] = res_emb[(size_t)node * DD + tid] * mC;
    if (tid < KK) {
        const int idx = edge_index[(size_t)node * KK + tid];
        const int v   = (idx != -1);
        const int si  = v ? idx : 0;
        sidx[tid]   = si;
        svalid[tid] = (float)v;
        snmask[tid] = mask[b * LL + si];
    }
    __syncthreads();

    if (tid == 0) {
        float s = 0.f;
#pragma unroll
        for (int r = 0; r < KK; ++r) s += svalid[r];
        snv = (s > 0.f) ? s : 1.f;
    }

    // central block: cols [0,128)
    for (int i = tid; i < KK * DD; i += 256) {
        const int r = i >> 7, c = i & 127;
        Xs[r * XS + c] = hc[c];
    }
    // neighbor block: cols [128,256)  (h = res * mask of the neighbor)
    for (int i = tid; i < KK * DD; i += 256) {
        const int r = i >> 7, c = i & 127;
        Xs[r * XS + 128 + c] = res_emb[(size_t)(b * LL + sidx[r]) * DD + c] * snmask[r];
    }
    // edge block: cols [256,384)
    const float* ebase = edge_emb + (size_t)node * KK * EE;
    for (int i = tid; i < KK * EE; i += 256) {
        const int r = i >> 7, c = i & 127;
        Xs[r * XS + 256 + c] = ebase[i];
    }
    __syncthreads();

    const int lane = tid & 31;
    const int wave = tid >> 5;

    mlp_layer(W1, b1, Xs,  XS, 2 * DD + EE, Ysh, YS, lane, wave);
    __syncthreads();
    mlp_layer(W2, b2, Ysh, YS, DD,          Xs,  XS, lane, wave);
    __syncthreads();
    mlp_layer(W3, b3, Xs,  XS, DD,          Ysh, YS, lane, wave);
    __syncthreads();

    // masked mean over K neighbors + residual
    if (tid < DD) {
        float s = 0.f;
#pragma unroll
        for (int r = 0; r < KK; ++r) s += Ysh[r * YS + tid] * svalid[r];
        const float out = (res_emb[(size_t)node * DD + tid] + s / snv) * mC;
        h_tmp[(size_t)node * DD + tid] = out;
    }
}

// ---------------------------------------------------------------------------
// Kernel 2: GraphNorm. One block per batch b, 128 threads (one per feature).
// ---------------------------------------------------------------------------
__global__ __launch_bounds__(128)
void graphnorm_kernel(const float* __restrict__ h_tmp,
                      const float* __restrict__ mask,
                      const float* __restrict__ gamma,
                      const float* __restrict__ beta,
                      float* __restrict__ h_out) {
    const int b = blockIdx.x;
    const int d = threadIdx.x;

    float s = 0.f, s2 = 0.f, n = 0.f;
    for (int l = 0; l < LL; ++l) {
        const float m = mask[b * LL + l];
        const float x = h_tmp[(size_t)(b * LL + l) * DD + d];
        s  += x * m;
        s2 += x * x * m;
        n  += m;
    }
    n = fmaxf(n, 1.0f);
    const float mean = s / n;
    const float var  = fmaxf(s2 / n - mean * mean, 0.0f);
    const float rstd = rsqrtf(var + 1e-5f);
    const float g = gamma[d], be = beta[d];

    for (int l = 0; l < LL; ++l) {
        const float m = mask[b * LL + l];
        const float x = h_tmp[(size_t)(b * LL + l) * DD + d];
        h_out[(size_t)(b * LL + l) * DD + d] = ((x - mean) * rstd * g + be) * m;
    }
}

// ---------------------------------------------------------------------------
// Kernel 3: edge MLP (384->128->128, gelu) on normalized h. One block / node.
// ---------------------------------------------------------------------------
__global__ __launch_bounds__(256)
void edge_kernel(const float* __restrict__ h_out,
                 const float* __restrict__ edge_emb,
                 const int*   __restrict__ edge_index,
                 const float* __restrict__ W1, const float* __restrict__ b1,
                 const float* __restrict__ W2, const float* __restrict__ b2,
                 float* __restrict__ new_edges) {
    __shared__ float Xs[KK * XS];
    __shared__ float Ysh[KK * YS];
    __shared__ float hc[DD];
    __shared__ int   sidx[KK];

    const int tid  = threadIdx.x;
    const int b    = blockIdx.x >> 9;
    const int l    = blockIdx.x & (LL - 1);
    const int node = b * LL + l;

    if (tid < DD) hc[tid] = h_out[(size_t)node * DD + tid];
    if (tid < KK) {
        const int idx = edge_index[(size_t)node * KK + tid];
        sidx[tid] = (idx != -1) ? idx : 0;
    }
    __syncthreads();

    for (int i = tid; i < KK * DD; i += 256) {
        const int r = i >> 7, c = i & 127;
        Xs[r * XS + c] = hc[c];
    }
    for (int i = tid; i < KK * DD; i += 256) {
        const int r = i >> 7, c = i & 127;
        Xs[r * XS + 128 + c] = h_out[(size_t)(b * LL + sidx[r]) * DD + c];
    }
    const float* ebase = edge_emb + (size_t)node * KK * EE;
    for (int i = tid; i < KK * EE; i += 256) {
        const int r = i >> 7, c = i & 127;
        Xs[r * XS + 256 + c] = ebase[i];
    }
    __syncthreads();

    const int lane = tid & 31;
    const int wave = tid >> 5;

    mlp_layer(W1, b1, Xs,  XS, 2 * DD + EE, Ysh, YS, lane, wave);
    __syncthreads();
    mlp_layer(W2, b2, Ysh, YS, DD,          Xs,  XS, lane, wave);
    __syncthreads();

    float* obase = new_edges + (size_t)node * KK * EE;
    for (int i = tid; i < KK * EE; i += 256) {
        const int r = i >> 7, c = i & 127;
        obase[i] = Xs[r * XS + c];
    }
}

// ---------------------------------------------------------------------------
// Launch. Input order (setup_inputs dict, lists flattened):
//  0 res_embedding  1 res_edge_embedding  2 edge_index  3 mask
//  4-6 msg_Ws       7-9 msg_bs            10-11 edge_Ws 12-13 edge_bs
//  14 gn_gamma      15 gn_beta
// Output: h [B,L,D] then new_edges [B,L,K,E], concatenated in d_out.
// ---------------------------------------------------------------------------
extern "C" void kernel_launch(void* const* d_in, const int* in_sizes, int n_in,
                              void* d_out, int out_size, void* d_ws, size_t ws_size,
                              hipStream_t stream) {
    const float* res_emb   = (const float*)d_in[0];
    const float* edge_emb  = (const float*)d_in[1];
    const int*   edge_idx  = (const int*)  d_in[2];
    const float* mask      = (const float*)d_in[3];
    const float* mW0 = (const float*)d_in[4];
    const float* mW1 = (const float*)d_in[5];
    const float* mW2 = (const float*)d_in[6];
    const float* mb0 = (const float*)d_in[7];
    const float* mb1 = (const float*)d_in[8];
    const float* mb2 = (const float*)d_in[9];
    const float* eW0 = (const float*)d_in[10];
    const float* eW1 = (const float*)d_in[11];
    const float* eb0 = (const float*)d_in[12];
    const float* eb1 = (const float*)d_in[13];
    const float* gng = (const float*)d_in[14];
    const float* gnb = (const float*)d_in[15];

    float* out       = (float*)d_out;
    float* h_out     = out;                              // [B,L,D]
    float* new_edges = out + (size_t)BB * LL * DD;       // [B,L,K,E]
    float* h_tmp     = (float*)d_ws;                     // [B,L,D] scratch (2 MB)

    const dim3 gridN(BB * LL);

    msg_kernel<<<gridN, dim3(256), 0, stream>>>(
        res_emb, edge_emb, edge_idx, mask,
        mW0, mb0, mW1, mb1, mW2, mb2, h_tmp);

    graphnorm_kernel<<<dim3(BB), dim3(DD), 0, stream>>>(
        h_tmp, mask, gng, gnb, h_out);

    edge_kernel<<<gridN, dim3(256), 0, stream>>>(
        h_out, edge_emb, edge_idx,
        eW0, eb0, eW1, eb1, new_edges);
}